// new_loss_model_84267258347529
// MI455X (gfx1250) — compile-verified
//
#include <hip/hip_runtime.h>
#include <math.h>

#define RR  2048
#define CC  1024
#define CHN 8
#define BAN 8
#define K64 64

typedef __attribute__((ext_vector_type(2))) float v2f;
typedef __attribute__((ext_vector_type(8))) float v8f;

// ---- d_out float layout (reference tuple, flattened in order) ----
#define OUT_P       0         // [R, CH, BA] = 131072
#define OUT_TOT     131072    // [8]
#define OUT_MAXNNZ  131080    // [8]
#define OUT_NUMCOL  131088    // [8]
#define OUT_NUMROW  131096    // [8]
#define OUT_NNZ     131104    // [CH, BA] = 64
#define OUT_COLDEN  131168    // [8]  (holds cdnum until k_final divides)
#define OUT_NRSEL   131176    // [1]
#define OUT_STATS_N 105       // floats from OUT_TOT to end

// ---- d_ws float layout ----
#define WS_LQPAD    0          // [R][16]  log(q) padded with zeros to N=16
#define WS_LQ       (RR * 16)  // [C][8]   log-domain accumulator

// -------------------------------------------------------------------------
__global__ void k_init(float* __restrict__ out, float* __restrict__ ws) {
    int t = blockIdx.x * blockDim.x + threadIdx.x;
    if (t < CC * CHN)    ws[WS_LQ + t]   = 0.0f;
    if (t < OUT_STATS_N) out[OUT_TOT + t] = 0.0f;
}

// -------------------------------------------------------------------------
// One wave32 per row: softmax over 64 elems (2 per lane), per-(ch) sums and
// (1-p) products, D row-sum (B128 vectorized), atomic global stats.
__global__ void k_row(const float* __restrict__ W, const float* __restrict__ D,
                      const float* __restrict__ g, float* __restrict__ out,
                      float* __restrict__ ws) {
    const int lane = threadIdx.x & 31;
    const int wave = threadIdx.x >> 5;
    const int r = blockIdx.x * (blockDim.x >> 5) + wave;
    if (r >= RR) return;

    float t0 = W[r * K64 + lane]      + g[r * K64 + lane];
    float t1 = W[r * K64 + 32 + lane] + g[r * K64 + 32 + lane];

    // wave max
    float m = fmaxf(t0, t1);
    for (int o = 16; o; o >>= 1) m = fmaxf(m, __shfl_xor(m, o, 32));
    float p0 = expf(t0 - m), p1 = expf(t1 - m);
    float sm = p0 + p1;
    for (int o = 16; o; o >>= 1) sm += __shfl_xor(sm, o, 32);
    float inv = 1.0f / sm;
    p0 *= inv; p1 *= inv;

    out[OUT_P + r * K64 + lane]      = p0;
    out[OUT_P + r * K64 + 32 + lane] = p1;

    // group-of-8 (per-ch) sum and product-of-(1-p), butterfly within octet
    float s0 = p0, s1 = p1, q0 = 1.0f - p0, q1 = 1.0f - p1;
    for (int o = 4; o; o >>= 1) {
        s0 += __shfl_xor(s0, o, 32);
        s1 += __shfl_xor(s1, o, 32);
        q0 *= __shfl_xor(q0, o, 32);
        q1 *= __shfl_xor(q1, o, 32);
    }

    // row max of P for num_row_sel
    float rmax = fmaxf(p0, p1);
    for (int o = 16; o; o >>= 1) rmax = fmaxf(rmax, __shfl_xor(rmax, o, 32));

    // row sum of D: vectorized B128 loads (512B per wave per instruction)
    const float4* D4 = (const float4*)(D + r * CC);
    float rd = 0.0f;
#pragma unroll
    for (int j = 0; j < CC / (32 * 4); ++j) {
        float4 v = D4[j * 32 + lane];
        rd += (v.x + v.y) + (v.z + v.w);
    }
    for (int o = 16; o; o >>= 1) rd += __shfl_xor(rd, o, 32);

    // nnz_ch_ba[ch*8+ba] += P * rowsumD  (element index == lane / lane+32)
    atomicAdd(&out[OUT_NNZ + lane],      p0 * rd);
    atomicAdd(&out[OUT_NNZ + 32 + lane], p1 * rd);

    if ((lane & 7) == 0) {
        int gi = lane >> 3;  // 0..3 ; p0 side -> ch=gi, p1 side -> ch=4+gi
        // clamp: avoid -inf logs -> WMMA 0*inf = NaN
        float lq0 = logf(fmaxf(q0, 1e-37f));
        float lq1 = logf(fmaxf(q1, 1e-37f));
        ws[WS_LQPAD + r * 16 + gi]      = lq0;
        ws[WS_LQPAD + r * 16 + 4 + gi]  = lq1;
        ws[WS_LQPAD + r * 16 + 8 + gi]  = 0.0f;   // pad cols 8..15
        ws[WS_LQPAD + r * 16 + 12 + gi] = 0.0f;
        atomicAdd(&out[OUT_NUMROW + gi],     s0);
        atomicAdd(&out[OUT_NUMROW + 4 + gi], s1);
        atomicAdd(&out[OUT_COLDEN + gi],     rd * s0);  // cdnum
        atomicAdd(&out[OUT_COLDEN + 4 + gi], rd * s1);
    }
    if (lane == 0) atomicAdd(&out[OUT_NRSEL], (rmax > 0.99f) ? 1.0f : 0.0f);
}

// -------------------------------------------------------------------------
// LQ = D^T @ lqpad via V_WMMA_F32_16X16X4_F32.
// Grid: (C/16 col tiles, 8 K-slices); one wave32 per block; EXEC all-1s
// through the WMMA loop. A = 16x4 tile of D^T (f32 layout: lanes 0-15 hold
// K=0,1; lanes 16-31 hold K=2,3). B = 4x16 of lqpad (rows striped likewise).
__global__ void k_wmma(const float* __restrict__ D, float* __restrict__ ws) {
    const int lane = threadIdx.x;            // blockDim.x == 32
    const int c0   = blockIdx.x * 16;
    const int rb   = blockIdx.y * (RR / 8);  // 256 rows per slice
    const int half = lane >> 4;              // 0: K=0,1 ; 1: K=2,3
    const int lid  = lane & 15;

    v8f acc = {};
    for (int k = 0; k < RR / 8; k += 4) {
        const int r0 = rb + k + 2 * half;
        v2f a, b;
        a.x = D[(r0 + 0) * CC + c0 + lid];               // A[m=lid, k]
        a.y = D[(r0 + 1) * CC + c0 + lid];
        b.x = ws[WS_LQPAD + (r0 + 0) * 16 + lid];        // B[k, n=lid]
        b.y = ws[WS_LQPAD + (r0 + 1) * 16 + lid];
        acc = __builtin_amdgcn_wmma_f32_16x16x4_f32(
            /*neg_a=*/false, a, /*neg_b=*/false, b,
            /*c_mod=*/(short)0, acc, /*reuse_a=*/false, /*reuse_b=*/false);
    }

    // D-matrix layout: VGPR j -> row M=j (lanes 0-15) / M=j+8 (lanes 16-31),
    // N = lane%16.  M = column-in-tile, N = ch (only 0..7 valid).
    if (lid < CHN) {
        const int cbase = c0 + half * 8;
#pragma unroll
        for (int j = 0; j < 8; ++j)
            atomicAdd(&ws[WS_LQ + (cbase + j) * CHN + lid], acc[j]);
    }
}

// -------------------------------------------------------------------------
// col_ch = 1 - exp(LQ); reduce over columns into num_col_ch.
__global__ void k_col(float* __restrict__ out, const float* __restrict__ ws) {
    const int t = blockIdx.x * blockDim.x + threadIdx.x;   // 0..8191
    const int lane = threadIdx.x & 31;
    float v = 1.0f - expf(ws[WS_LQ + t]);
    // lanes {L, L+8, L+16, L+24} share the same ch = t & 7
    v += __shfl_xor(v, 8, 32);
    v += __shfl_xor(v, 16, 32);
    if (lane < 8) atomicAdd(&out[OUT_NUMCOL + (t & 7)], v);
}

// -------------------------------------------------------------------------
__global__ void k_final(float* __restrict__ out) {
    const int t = threadIdx.x;
    if (t < CHN) {
        float mx = out[OUT_NNZ + t * BAN];
        for (int b = 1; b < BAN; ++b)
            mx = fmaxf(mx, out[OUT_NNZ + t * BAN + b]);
        out[OUT_MAXNNZ + t] = mx;
        const float ncol = out[OUT_NUMCOL + t];
        const float nrow = out[OUT_NUMROW + t];
        out[OUT_TOT + t] = mx + ncol + nrow;
        out[OUT_COLDEN + t] = out[OUT_COLDEN + t] / nrow / ncol;
    }
}

// -------------------------------------------------------------------------
extern "C" void kernel_launch(void* const* d_in, const int* in_sizes, int n_in,
                              void* d_out, int out_size, void* d_ws, size_t ws_size,
                              hipStream_t stream) {
    const float* W = (const float*)d_in[0];
    const float* D = (const float*)d_in[1];
    const float* g = (const float*)d_in[2];
    // d_in[3] = i : has no effect on the math (temperature quirk)
    float* out = (float*)d_out;
    float* ws  = (float*)d_ws;

    k_init<<<32, 256, 0, stream>>>(out, ws);
    k_row<<<RR / 8, 256, 0, stream>>>(W, D, g, out, ws);
    dim3 g2(CC / 16, 8);
    k_wmma<<<g2, 32, 0, stream>>>(D, ws);
    k_col<<<(CC * CHN) / 256, 256, 0, stream>>>(out, ws);
    k_final<<<1, 64, 0, stream>>>(out);
}